// GroupProjection_25228637897402
// MI455X (gfx1250) — compile-verified
//
#include <hip/hip_runtime.h>
#include <hip/hip_bf16.h>

typedef __attribute__((ext_vector_type(2))) float v2f;
typedef __attribute__((ext_vector_type(8))) float v8f;

#define BDIM   256
#define TDIM   512
#define FDIM   256
#define GN     8
#define GFN    32
#define GON    64
#define DMODEL (GN * GON)     // 512
#define BT     (BDIM * TDIM)  // 131072 rows
#define ROWS   16             // rows of X per block
#define XS_STRIDE 260         // 260 mod 64 = 4 -> conflict-free A-frag reads

__global__ __launch_bounds__(256)
void GroupProjection_wmma_f32_kernel(const float* __restrict__ x,
                                     const float* __restrict__ W,
                                     const float* __restrict__ bias,
                                     const int*   __restrict__ idx,
                                     float*       __restrict__ out)
{
    __shared__ float xs[ROWS * XS_STRIDE];

    const int row0 = blockIdx.x * ROWS;
    const int tid  = threadIdx.x;

    // ---- Cooperative gathered load of the A tile: 16 rows x 256 features ----
    // linear = tid + i*256 keeps consecutive lanes on consecutive columns
    // (coalesced for the identity idx in the reference).
#pragma unroll
    for (int i = 0; i < (ROWS * FDIM) / 256; ++i) {
        int linear = tid + i * 256;
        int r = linear >> 8;          // / FDIM
        int c = linear & (FDIM - 1);  // % FDIM
        xs[r * XS_STRIDE + c] = x[(size_t)(row0 + r) * FDIM + idx[c]];
    }

    const int lane = tid & 31;
    const int g    = tid >> 5;   // wave id == group id (8 waves/block)
    const int half = lane >> 4;  // 0: lanes 0-15, 1: lanes 16-31
    const int mn   = lane & 15;  // M for A, N for B/C/D

    // ---- Preload this group's full 32x64 W tile as B fragments ----
    // B layout for V_WMMA_F32_16X16X4_F32 (4x16 KxN):
    //   VGPR v, lane l -> B[k0 + 2*(l>>4) + v][l&15]
    v2f bf[8][4];
    const float* Wg = W + (size_t)g * GFN * GON;
#pragma unroll
    for (int ks = 0; ks < 8; ++ks) {
        int k0 = ks * 4 + 2 * half;
#pragma unroll
        for (int ct = 0; ct < 4; ++ct) {
            int col = ct * 16 + mn;
            v2f b;
            b.x = Wg[(k0 + 0) * GON + col];
            b.y = Wg[(k0 + 1) * GON + col];
            bf[ks][ct] = b;
        }
    }

    // ---- Init accumulators with the bias: D = A*B + C folds "+b" in ----
    // C/D layout: VGPR j, lane l -> D[j + 8*(l>>4)][l&15]; bias depends only on N.
    v8f acc[4];
#pragma unroll
    for (int ct = 0; ct < 4; ++ct) {
        float bv = bias[g * GON + ct * 16 + mn];
        v8f c;
#pragma unroll
        for (int j = 0; j < 8; ++j) c[j] = bv;
        acc[ct] = c;
    }

    __syncthreads();

    // ---- K loop: 8 steps of K=4, A fragment reused across 4 col-tiles ----
    // A layout (16x4 MxK): VGPR v, lane l -> A[l&15][k0 + 2*(l>>4) + v]
#pragma unroll
    for (int ks = 0; ks < 8; ++ks) {
        const float* ap = &xs[mn * XS_STRIDE + g * GFN + ks * 4 + 2 * half];
        v2f a;
        a.x = ap[0];
        a.y = ap[1];
#pragma unroll
        for (int ct = 0; ct < 4; ++ct) {
            acc[ct] = __builtin_amdgcn_wmma_f32_16x16x4_f32(
                /*neg_a=*/false, a, /*neg_b=*/false, bf[ks][ct],
                /*c_mod=*/(short)0, acc[ct],
                /*reuse_a=*/false, /*reuse_b=*/false);
        }
    }

    // ---- Store D tiles: wave g covers out cols [g*64, g*64+64) ----
#pragma unroll
    for (int ct = 0; ct < 4; ++ct) {
#pragma unroll
        for (int j = 0; j < 8; ++j) {
            int row = row0 + j + 8 * half;
            out[(size_t)row * DMODEL + g * GON + ct * 16 + mn] = acc[ct][j];
        }
    }
}

extern "C" void kernel_launch(void* const* d_in, const int* in_sizes, int n_in,
                              void* d_out, int out_size, void* d_ws, size_t ws_size,
                              hipStream_t stream) {
    (void)in_sizes; (void)n_in; (void)out_size; (void)d_ws; (void)ws_size;
    const float* x    = (const float*)d_in[0];
    const float* W    = (const float*)d_in[1];
    const float* bias = (const float*)d_in[2];
    const int*   idx  = (const int*)d_in[3];
    float*       out  = (float*)d_out;

    dim3 grid(BT / ROWS);   // 8192 blocks
    dim3 block(256);        // 8 wave32 waves; wave id == group id
    GroupProjection_wmma_f32_kernel<<<grid, block, 0, stream>>>(x, W, bias, idx, out);
}